// GcnLayer_12120397709394
// MI455X (gfx1250) — compile-verified
//
#include <hip/hip_runtime.h>
#include <hip/hip_bf16.h>

// ---------------------------------------------------------------------------
// GCN layer for MI455X (gfx1250, wave32):
//   deg = segsum(edge_val by row) + 1 ; inv = rsqrt(deg)
//   pooled = A @ (inv*features) + inv*(inv*features)        [N,128]
//   out = relu(pooled @ W)                                  [N,256]
// Edge phase is L2/atomic bound; GEMM uses exact-precision f32 WMMA
// (V_WMMA_F32_16X16X4_F32) since matrix math is nowhere near the bottleneck.
// ---------------------------------------------------------------------------

typedef __attribute__((ext_vector_type(2))) float v2f;
typedef __attribute__((ext_vector_type(8))) float v8f;

#define D_FEAT 128
#define UNITS  256

// ---- Pass 1: degree init (self loop contributes +1) ------------------------
__global__ __launch_bounds__(256) void k_init_deg(float* __restrict__ deg, int n) {
    int i = blockIdx.x * blockDim.x + threadIdx.x;
    if (i < n) deg[i] = 1.0f;
}

// ---- Pass 2: degree accumulation over edges --------------------------------
__global__ __launch_bounds__(256) void k_accum_deg(const int* __restrict__ erow,
                                                   const float* __restrict__ eval,
                                                   float* __restrict__ deg, int E) {
    int e = blockIdx.x * blockDim.x + threadIdx.x;
    if (e < E) atomicAdd(&deg[erow[e]], eval[e]);
}

// ---- Pass 3: inv = rsqrt(deg); pooled = inv^2 * features (self-loop term) --
// One thread per (node, float4 group): 32 threads cover a 128-feature row.
__global__ __launch_bounds__(256) void k_normalize(const float* __restrict__ feat,
                                                   const float* __restrict__ deg,
                                                   float* __restrict__ inv,
                                                   float* __restrict__ pooled, int n) {
    int tid = blockIdx.x * blockDim.x + threadIdx.x;
    if (tid >= n * 32) return;
    int node = tid >> 5;
    int g    = tid & 31;
    float iv = rsqrtf(deg[node]);
    if (g == 0) inv[node] = iv;
    float s = iv * iv;                       // inv * (inv * feature)
    const float4 f = *(const float4*)(feat + node * D_FEAT + g * 4);
    float4 o; o.x = s * f.x; o.y = s * f.y; o.z = s * f.z; o.w = s * f.w;
    *(float4*)(pooled + node * D_FEAT + g * 4) = o;
}

// ---- Pass 4: edge gather/scale/scatter-add ---------------------------------
// One wave (32 lanes) per edge; lane handles 4 contiguous features (float4
// gather = coalesced 512B row read; 4 f32 atomics into L2-resident pooled).
__global__ __launch_bounds__(256) void k_scatter(const int* __restrict__ erow,
                                                 const int* __restrict__ ecol,
                                                 const float* __restrict__ eval,
                                                 const float* __restrict__ feat,
                                                 const float* __restrict__ inv,
                                                 float* __restrict__ pooled, int E) {
    int tid = blockIdx.x * blockDim.x + threadIdx.x;
    int e = tid >> 5;
    if (e >= E) return;
    int lane = tid & 31;
    int r = erow[e];
    int c = ecol[e];
    float s = eval[e] * inv[c];              // edge_val * invsqrt_deg[col]
    const float4 f = *(const float4*)(feat + c * D_FEAT + lane * 4);
    float* p = pooled + r * D_FEAT + lane * 4;
    atomicAdd(p + 0, s * f.x);
    atomicAdd(p + 1, s * f.y);
    atomicAdd(p + 2, s * f.z);
    atomicAdd(p + 3, s * f.w);
}

// ---- Pass 5: out = relu(pooled @ W) via V_WMMA_F32_16X16X4_F32 -------------
// Each wave owns one 16-column tile of W: full B panel (128x16 f32) lives in
// 64 VGPRs (32 x v2f), then grid-strides over 16-row strips of pooled.
// A 16x4 f32 layout : lane L -> row M=L%16, VGPR0 = K=(L/16)*2, VGPR1 = K+1.
// B 4x16  f32 layout: lane L -> col N=L%16, VGPR0 = K=(L/16)*2, VGPR1 = K+1.
// C/D 16x16 f32     : VGPR i, lane L -> M = i + (L/16)*8, N = L%16.
__global__ __launch_bounds__(256) void k_gemm_wmma(const float* __restrict__ P,
                                                   const float* __restrict__ W,
                                                   float* __restrict__ out, int M) {
    const int lane   = threadIdx.x & 31;
    const int gwave  = blockIdx.x * (blockDim.x >> 5) + (threadIdx.x >> 5);
    const int nwaves = gridDim.x * (blockDim.x >> 5);

    const int ntile  = gwave & 15;           // 16 tiles of 16 columns = 256
    const int wgroup = gwave >> 4;           // strip-group this wave walks
    const int ngroup = nwaves >> 4;

    const int n0  = ntile * 16;
    const int row = lane & 15;               // M (for A) / N (for B,C/D)
    const int kb  = (lane >> 4) << 1;        // K sub-offset: 0 or 2

    // Stage the whole B panel for this column tile in registers (read W once).
    v2f b[32];
#pragma unroll
    for (int ks = 0; ks < 32; ++ks) {
        const int kk = ks * 4 + kb;
        b[ks].x = W[kk * UNITS + n0 + row];
        b[ks].y = W[(kk + 1) * UNITS + n0 + row];
    }

    const int nstrips = M >> 4;              // 50000/16 = 3125 (exact)
    for (int strip = wgroup; strip < nstrips; strip += ngroup) {
        const int m0 = strip << 4;
        v8f acc = {};
        const float* arow = P + (m0 + row) * D_FEAT + kb;
#pragma unroll
        for (int ks = 0; ks < 32; ++ks) {
            v2f a;
            a.x = arow[ks * 4 + 0];          // contiguous pair -> b64 load
            a.y = arow[ks * 4 + 1];
            acc = __builtin_amdgcn_wmma_f32_16x16x4_f32(
                /*neg_a=*/false, a, /*neg_b=*/false, b[ks],
                /*c_mod=*/(short)0, acc, /*reuse_a=*/false, /*reuse_b=*/false);
        }
#pragma unroll
        for (int i = 0; i < 8; ++i) {
            const int m = m0 + i + ((lane >> 4) << 3);
            out[m * UNITS + n0 + row] = fmaxf(acc[i], 0.0f);  // ReLU
        }
    }
}

// ---------------------------------------------------------------------------
extern "C" void kernel_launch(void* const* d_in, const int* in_sizes, int n_in,
                              void* d_out, int out_size, void* d_ws, size_t ws_size,
                              hipStream_t stream) {
    const float* feat = (const float*)d_in[0];   // [N,128]
    const int*   erow = (const int*)  d_in[1];   // [E]
    const int*   ecol = (const int*)  d_in[2];   // [E]
    const float* eval = (const float*)d_in[3];   // [E]
    const float* W    = (const float*)d_in[4];   // [128,256]
    float*       out  = (float*)d_out;           // [N,256]

    const int N = in_sizes[0] / D_FEAT;          // 50000
    const int E = in_sizes[1];                   // 800000

    // Workspace layout (all re-initialized every call => deterministic):
    //   deg   : N floats
    //   inv   : N floats
    //   pooled: N*128 floats
    float* deg    = (float*)d_ws;
    float* inv    = deg + N;
    float* pooled = inv + N;

    const int T = 256;
    k_init_deg <<<(N + T - 1) / T, T, 0, stream>>>(deg, N);
    k_accum_deg<<<(E + T - 1) / T, T, 0, stream>>>(erow, eval, deg, E);
    k_normalize<<<(N * 32 + T - 1) / T, T, 0, stream>>>(feat, deg, inv, pooled, N);
    k_scatter  <<<(E * 32 + T - 1) / T, T, 0, stream>>>(erow, ecol, eval, feat, inv,
                                                        pooled, E);
    // 512 blocks * 8 waves = 4096 waves = 16 col-tiles x 256 strip-groups.
    k_gemm_wmma<<<512, T, 0, stream>>>(pooled, W, out, N);
}